// AttentionBlock_33569464385965
// MI455X (gfx1250) — compile-verified
//
#include <hip/hip_runtime.h>

#define B_  8
#define C_  256
#define CQ_ 8
#define H_  128
#define W_  128
#define HD_ 32
#define WD_ 32
#define N_  1024   // HD_*WD_

typedef float v2f __attribute__((ext_vector_type(2)));
typedef float v8f __attribute__((ext_vector_type(8)));

// ---------------------------------------------------------------------------
// 1) 4x4 average pool: x[B,C,128,128] -> xf[B,C,1024]  (memory-bound, float4)
// ---------------------------------------------------------------------------
__global__ void pool_kernel(const float* __restrict__ x, float* __restrict__ xf) {
    int idx = blockIdx.x * blockDim.x + threadIdx.x;      // B*C*N threads
    int n   = idx & (N_ - 1);
    int bc  = idx >> 10;                                  // b*C + c
    int hd  = n >> 5, wd = n & 31;
    const float* xp = x + ((size_t)bc * H_ + hd * 4) * W_ + wd * 4;
    float s = 0.f;
#pragma unroll
    for (int i = 0; i < 4; ++i) {
        float4 v = *reinterpret_cast<const float4*>(xp + i * W_);
        s += v.x + v.y + v.z + v.w;
    }
    xf[idx] = s * 0.0625f;
}

// ---------------------------------------------------------------------------
// 2a) q/k projection packed into one 16-row tile: rows 0-7 Wq, rows 8-15 Wk
// ---------------------------------------------------------------------------
__global__ void proj_qk_kernel(const float* __restrict__ Wq, const float* __restrict__ bq,
                               const float* __restrict__ Wk, const float* __restrict__ bk,
                               const float* __restrict__ xf,
                               float* __restrict__ qout, float* __restrict__ kout) {
    int wave = (blockIdx.x * blockDim.x + threadIdx.x) >> 5;
    int lane = threadIdx.x & 31;
    int b     = wave >> 6;             // / 64 n-tiles
    int nbase = (wave & 63) << 4;
    int j     = lane & 15;             // A row (m) and B col (n) index
    int koff  = (lane >> 4) << 1;      // K sub-offset per half-wave
    const float* Arow = (j < 8) ? (Wq + (size_t)j * C_) : (Wk + (size_t)(j - 8) * C_);
    const float* xfb  = xf + (size_t)b * C_ * N_;
    v8f acc = {};
    for (int kb = 0; kb < C_; kb += 4) {
        int k0 = kb + koff;
        v2f a = *reinterpret_cast<const v2f*>(Arow + k0);
        v2f bm;
        bm.x = xfb[(size_t)k0 * N_ + nbase + j];
        bm.y = xfb[(size_t)(k0 + 1) * N_ + nbase + j];
        acc = __builtin_amdgcn_wmma_f32_16x16x4_f32(false, a, false, bm, (short)0, acc, false, false);
    }
    int mh = (lane >> 4) << 3;
#pragma unroll
    for (int rr = 0; rr < 8; ++rr) {
        int row = mh + rr;
        float val = acc[rr];
        if (row < 8)
            qout[((size_t)b * CQ_ + row) * N_ + nbase + j] = val + bq[row];
        else
            kout[((size_t)b * CQ_ + row - 8) * N_ + nbase + j] = val + bk[row - 8];
    }
}

// ---------------------------------------------------------------------------
// 2b) v projection: v[b] = Wv[256,256] @ xf[b] + bv ; one 16x16 tile per wave
// ---------------------------------------------------------------------------
__global__ void proj_v_kernel(const float* __restrict__ Wv, const float* __restrict__ bv,
                              const float* __restrict__ xf, float* __restrict__ vout) {
    int wave = (blockIdx.x * blockDim.x + threadIdx.x) >> 5;
    int lane = threadIdx.x & 31;
    int b     = wave >> 10;            // / (16*64)
    int r     = wave & 1023;
    int obase = (r >> 6) << 4;
    int nbase = (r & 63) << 4;
    int j     = lane & 15;
    int koff  = (lane >> 4) << 1;
    const float* xfb = xf + (size_t)b * C_ * N_;
    v8f acc = {};
    for (int kb = 0; kb < C_; kb += 4) {
        int k0 = kb + koff;
        v2f a = *reinterpret_cast<const v2f*>(Wv + (size_t)(obase + j) * C_ + k0);
        v2f bm;
        bm.x = xfb[(size_t)k0 * N_ + nbase + j];
        bm.y = xfb[(size_t)(k0 + 1) * N_ + nbase + j];
        acc = __builtin_amdgcn_wmma_f32_16x16x4_f32(false, a, false, bm, (short)0, acc, false, false);
    }
    float* ob = vout + (size_t)b * C_ * N_;
    int mh = (lane >> 4) << 3;
#pragma unroll
    for (int rr = 0; rr < 8; ++rr) {
        int row = obase + mh + rr;
        ob[(size_t)row * N_ + nbase + j] = acc[rr] + bv[row];
    }
}

// ---------------------------------------------------------------------------
// 3) scores S[b,m,n] = sum_c q[b,c,m]*k[b,c,n]; K=CQ=8 -> two WMMAs per tile
// ---------------------------------------------------------------------------
__global__ void scores_kernel(const float* __restrict__ q, const float* __restrict__ k,
                              float* __restrict__ S) {
    int wave = (blockIdx.x * blockDim.x + threadIdx.x) >> 5;
    int lane = threadIdx.x & 31;
    int b     = wave >> 12;            // / 4096
    int r     = wave & 4095;
    int mbase = (r >> 6) << 4;
    int nbase = (r & 63) << 4;
    int j     = lane & 15;
    int koff  = (lane >> 4) << 1;
    const float* qb = q + (size_t)b * CQ_ * N_;
    const float* kp = k + (size_t)b * CQ_ * N_;
    v8f acc = {};
#pragma unroll
    for (int cb = 0; cb < CQ_; cb += 4) {
        int c0 = cb + koff;
        v2f a, bm;
        a.x  = qb[(size_t)c0 * N_ + mbase + j];
        a.y  = qb[(size_t)(c0 + 1) * N_ + mbase + j];
        bm.x = kp[(size_t)c0 * N_ + nbase + j];
        bm.y = kp[(size_t)(c0 + 1) * N_ + nbase + j];
        acc = __builtin_amdgcn_wmma_f32_16x16x4_f32(false, a, false, bm, (short)0, acc, false, false);
    }
    float* Sb = S + (size_t)b * N_ * N_;
    int mh = (lane >> 4) << 3;
#pragma unroll
    for (int rr = 0; rr < 8; ++rr)
        Sb[(size_t)(mbase + mh + rr) * N_ + nbase + j] = acc[rr];
}

// ---------------------------------------------------------------------------
// 4) row softmax over n (1024), one wave32 per row, float4 I/O, shfl reduce
// ---------------------------------------------------------------------------
__global__ void softmax_kernel(float* __restrict__ S) {
    int row  = (blockIdx.x * blockDim.x + threadIdx.x) >> 5;  // b*N + m
    int lane = threadIdx.x & 31;
    float* p = S + (size_t)row * N_;
    float4 vals[8];
    float mx = -3.402823466e+38f;
#pragma unroll
    for (int i = 0; i < 8; ++i) {
        vals[i] = *reinterpret_cast<const float4*>(p + (((i << 5) + lane) << 2));
        mx = fmaxf(mx, fmaxf(fmaxf(vals[i].x, vals[i].y), fmaxf(vals[i].z, vals[i].w)));
    }
#pragma unroll
    for (int off = 16; off > 0; off >>= 1)
        mx = fmaxf(mx, __shfl_xor(mx, off, 32));
    float sum = 0.f;
#pragma unroll
    for (int i = 0; i < 8; ++i) {
        vals[i].x = __expf(vals[i].x - mx);
        vals[i].y = __expf(vals[i].y - mx);
        vals[i].z = __expf(vals[i].z - mx);
        vals[i].w = __expf(vals[i].w - mx);
        sum += vals[i].x + vals[i].y + vals[i].z + vals[i].w;
    }
#pragma unroll
    for (int off = 16; off > 0; off >>= 1)
        sum += __shfl_xor(sum, off, 32);
    float inv = 1.0f / sum;
#pragma unroll
    for (int i = 0; i < 8; ++i) {
        vals[i].x *= inv; vals[i].y *= inv; vals[i].z *= inv; vals[i].w *= inv;
        *reinterpret_cast<float4*>(p + (((i << 5) + lane) << 2)) = vals[i];
    }
}

// ---------------------------------------------------------------------------
// 5) out[b,c,m] = sum_n v[b,c,n]*attn[b,m,n]
//    LDS-tiled workgroup GEMM: 512 threads = 16 waves computing a 64x64 tile
//    (4x4 grid of 16x16 WMMA sub-tiles). K-panels of 64 staged in LDS with
//    register double-buffering: global loads for chunk kc+1 are issued before
//    the 16 WMMAs of chunk kc, so L2 latency overlaps matrix compute.
//    LDS rows padded to 68 floats -> conflict-free b64 fragment reads.
// ---------------------------------------------------------------------------
#define KC_   64          // K-chunk depth
#define LDP_  68          // padded LDS row stride (floats)

__global__ __launch_bounds__(512)
void outgemm_lds_kernel(const float* __restrict__ v, const float* __restrict__ attn,
                        float* __restrict__ o) {
    __shared__ float Als[2][64][LDP_];
    __shared__ float Bls[2][64][LDP_];

    int tid  = threadIdx.x;
    int lane = tid & 31;
    int wave = tid >> 5;                 // 0..15
    int wc   = wave >> 2;                // c sub-tile 0..3
    int wm   = wave & 3;                 // m sub-tile 0..3

    int blk = blockIdx.x;                // B * 4 * 16
    int b   = blk >> 6;
    int r   = blk & 63;
    int c0  = (r >> 4) << 6;             // c tile base (x64)
    int m0  = (r & 15) << 6;             // m tile base (x64)

    const float* vb = v    + (size_t)b * C_ * N_;
    const float* ab = attn + (size_t)b * N_ * N_;

    // cooperative load mapping: each thread moves 2 float4 per panel per chunk
    int col4 = (tid & 15) << 2;          // 0..60
    int row  = tid >> 4;                 // 0..31 (and row+32)
    const float* gA0 = vb + (size_t)(c0 + row)      * N_ + col4;
    const float* gA1 = vb + (size_t)(c0 + row + 32) * N_ + col4;
    const float* gB0 = ab + (size_t)(m0 + row)      * N_ + col4;
    const float* gB1 = ab + (size_t)(m0 + row + 32) * N_ + col4;

    int j    = lane & 15;                // fragment row/col index
    int koff = (lane >> 4) << 1;         // K sub-offset per half-wave
    int mh   = (lane >> 4) << 3;         // C/D M offset per half-wave

    // preload chunk 0 into LDS buffer 0
    float4 ra0 = *reinterpret_cast<const float4*>(gA0);
    float4 ra1 = *reinterpret_cast<const float4*>(gA1);
    float4 rb0 = *reinterpret_cast<const float4*>(gB0);
    float4 rb1 = *reinterpret_cast<const float4*>(gB1);
    *reinterpret_cast<float4*>(&Als[0][row][col4])      = ra0;
    *reinterpret_cast<float4*>(&Als[0][row + 32][col4]) = ra1;
    *reinterpret_cast<float4*>(&Bls[0][row][col4])      = rb0;
    *reinterpret_cast<float4*>(&Bls[0][row + 32][col4]) = rb1;
    __syncthreads();

    v8f acc = {};
#pragma unroll 1
    for (int kc = 0; kc < N_ / KC_; ++kc) {
        int buf = kc & 1;
        if (kc < (N_ / KC_) - 1) {
            int go = (kc + 1) * KC_;
            ra0 = *reinterpret_cast<const float4*>(gA0 + go);
            ra1 = *reinterpret_cast<const float4*>(gA1 + go);
            rb0 = *reinterpret_cast<const float4*>(gB0 + go);
            rb1 = *reinterpret_cast<const float4*>(gB1 + go);
        }
        if (kc < (N_ / KC_) - 2) {
            int po = (kc + 2) * KC_;
            __builtin_prefetch(gA0 + po, 0, 0);
            __builtin_prefetch(gA1 + po, 0, 0);
            __builtin_prefetch(gB0 + po, 0, 0);
            __builtin_prefetch(gB1 + po, 0, 0);
        }
        // 16 WMMAs over the staged 64-deep K-chunk
#pragma unroll
        for (int ks = 0; ks < KC_; ks += 4) {
            v2f a  = *reinterpret_cast<const v2f*>(&Als[buf][(wc << 4) + j][ks + koff]);
            v2f bm = *reinterpret_cast<const v2f*>(&Bls[buf][(wm << 4) + j][ks + koff]);
            acc = __builtin_amdgcn_wmma_f32_16x16x4_f32(false, a, false, bm, (short)0, acc, false, false);
        }
        if (kc < (N_ / KC_) - 1) {
            int nb = buf ^ 1;
            *reinterpret_cast<float4*>(&Als[nb][row][col4])      = ra0;
            *reinterpret_cast<float4*>(&Als[nb][row + 32][col4]) = ra1;
            *reinterpret_cast<float4*>(&Bls[nb][row][col4])      = rb0;
            *reinterpret_cast<float4*>(&Bls[nb][row + 32][col4]) = rb1;
            __syncthreads();
        }
    }

    float* ob = o + (size_t)b * C_ * N_;
#pragma unroll
    for (int rr = 0; rr < 8; ++rr)
        ob[(size_t)(c0 + (wc << 4) + mh + rr) * N_ + m0 + (wm << 4) + j] = acc[rr];
}

// ---------------------------------------------------------------------------
// 6) bilinear x4 upsample (half-pixel, edge clamp) + y = gamma*out + x
// ---------------------------------------------------------------------------
__global__ void upsample_residual_kernel(const float* __restrict__ o, const float* __restrict__ x,
                                         const float* __restrict__ gamma, float* __restrict__ y) {
    int idx = blockIdx.x * blockDim.x + threadIdx.x;      // B*C*H*W
    int w  = idx & (W_ - 1);
    int h  = (idx >> 7) & (H_ - 1);
    int bc = idx >> 14;
    float sy = (h + 0.5f) * 0.25f - 0.5f;
    float sx = (w + 0.5f) * 0.25f - 0.5f;
    int y0 = (int)floorf(sy); float ty = sy - (float)y0;
    int x0 = (int)floorf(sx); float tx = sx - (float)x0;
    int i0 = min(max(y0, 0), HD_ - 1), i1 = min(max(y0 + 1, 0), HD_ - 1);
    int j0 = min(max(x0, 0), WD_ - 1), j1 = min(max(x0 + 1, 0), WD_ - 1);
    const float* ob = o + (size_t)bc * N_;
    float v00 = ob[i0 * WD_ + j0], v01 = ob[i0 * WD_ + j1];
    float v10 = ob[i1 * WD_ + j0], v11 = ob[i1 * WD_ + j1];
    float val = (1.f - ty) * ((1.f - tx) * v00 + tx * v01)
              +        ty  * ((1.f - tx) * v10 + tx * v11);
    y[idx] = gamma[0] * val + x[idx];
}

// ---------------------------------------------------------------------------
extern "C" void kernel_launch(void* const* d_in, const int* in_sizes, int n_in,
                              void* d_out, int out_size, void* d_ws, size_t ws_size,
                              hipStream_t stream) {
    (void)in_sizes; (void)n_in; (void)out_size; (void)ws_size;
    const float* x     = (const float*)d_in[0];
    const float* Wq    = (const float*)d_in[1];
    const float* bq    = (const float*)d_in[2];
    const float* Wk    = (const float*)d_in[3];
    const float* bk    = (const float*)d_in[4];
    const float* Wv    = (const float*)d_in[5];
    const float* bv    = (const float*)d_in[6];
    const float* gamma = (const float*)d_in[7];
    float* y  = (float*)d_out;

    float* ws = (float*)d_ws;
    float* xf = ws;                                 // B*C*N   (2M floats)
    float* q  = xf + (size_t)B_ * C_ * N_;          // B*CQ*N
    float* k  = q  + (size_t)B_ * CQ_ * N_;         // B*CQ*N
    float* v  = k  + (size_t)B_ * CQ_ * N_;         // B*C*N   (2M floats)
    float* S  = v  + (size_t)B_ * C_ * N_;          // B*N*N   (8M floats)

    pool_kernel<<<(B_ * C_ * N_) / 256, 256, 0, stream>>>(x, xf);
    proj_qk_kernel<<<(B_ * 64) / 4, 128, 0, stream>>>(Wq, bq, Wk, bk, xf, q, k);
    proj_v_kernel<<<(B_ * 16 * 64) / 4, 128, 0, stream>>>(Wv, bv, xf, v);
    scores_kernel<<<(B_ * 64 * 64) / 4, 128, 0, stream>>>(q, k, S);
    softmax_kernel<<<(B_ * N_) / 8, 256, 0, stream>>>(S);
    // xf is dead from here on: reuse it as out_small [B,C,N]
    outgemm_lds_kernel<<<B_ * 4 * 16, 512, 0, stream>>>(v, S, xf);
    upsample_residual_kernel<<<(B_ * C_ * H_ * W_) / 256, 256, 0, stream>>>(xf, x, gamma, y);
}